// multi_head_attention_40192303956289
// MI455X (gfx1250) — compile-verified
//
#include <hip/hip_runtime.h>
#include <stdint.h>

typedef __attribute__((ext_vector_type(2))) float v2f;
typedef __attribute__((ext_vector_type(4))) float v4f;
typedef __attribute__((ext_vector_type(8))) float v8f;

#define E_DIM 1024
#define W_DIM 1024
#define B_DIM 8
#define H_DIM 16
#define DH_DIM 64

__device__ __forceinline__ v8f wmma_f32(v2f a, v2f b, v8f c) {
  // V_WMMA_F32_16X16X4_F32 : D = A(16x4) * B(4x16) + C(16x16)
  return __builtin_amdgcn_wmma_f32_16x16x4_f32(
      /*neg_a=*/false, a, /*neg_b=*/false, b,
      /*c_mod=*/(short)0, c, /*reuse_a=*/false, /*reuse_b=*/false);
}

// Async memory -> LDS copy, 16 bytes per lane, tracked by ASYNCcnt.
__device__ __forceinline__ void async_b128(const float* gsrc, const float* lds_dst) {
  uint32_t l = (uint32_t)(uintptr_t)lds_dst;  // low 32 bits = LDS offset
  asm volatile("global_load_async_to_lds_b128 %0, %1, off"
               :: "v"(l), "v"(gsrc) : "memory");
}
__device__ __forceinline__ void async_wait3() {
  asm volatile("s_wait_asynccnt 3" ::: "memory");
}
__device__ __forceinline__ void async_wait0() {
  asm volatile("s_wait_asynccnt 0" ::: "memory");
}

// ---------------------------------------------------------------------------
// Generic fp32 GEMM: C[b] = Wm(1024x1024) @ X[b](1024x1024) (+ bias[e])
// grid = (W/128, E/64, batch), block = 256 threads (8 waves, 2m x 4n)
// Workgroup tile 64x128; each wave a 32x32 tile (2x2 WMMA accumulators).
// K staged in 16-wide chunks, double-buffered via GLOBAL_LOAD_ASYNC_TO_LDS.
// ---------------------------------------------------------------------------
#define AP 20    // A tile pitch (floats): 80B rows -> b128-aligned
#define BP 132   // B tile pitch (floats): 528B rows -> b128-aligned
#define NCHUNK (E_DIM / 16)

__global__ __launch_bounds__(256)
void gemm_f32_kernel(const float* __restrict__ Wm,
                     const float* __restrict__ X,
                     float* __restrict__ C,
                     const float* __restrict__ bias) {
  __shared__ float At[2][64][AP];   // [buf][m 64][k 16]
  __shared__ float Bt[2][16][BP];   // [buf][k 16][n 128]

  const int tid  = threadIdx.x;
  const int lane = tid & 31;
  const int wave = tid >> 5;
  const int hl   = lane & 15;   // lane within half-wave
  const int hi   = lane >> 4;   // half-wave index (0/1)
  const int mw   = wave >> 2;   // wave m position (0..1)
  const int nw   = wave & 3;    // wave n position (0..3)

  const int b  = blockIdx.z;
  const int m0 = blockIdx.y * 64;
  const int n0 = blockIdx.x * 128;

  const float* Xb = X + (size_t)b * E_DIM * W_DIM;
  float*       Cb = C + (size_t)b * E_DIM * W_DIM;

  // Per-thread staging coordinates (float4 granularity)
  const int arow = tid >> 2, ac4 = (tid & 3) * 4;          // A: 64x16
  const int br0  = tid >> 5, bc0 = (tid & 31) * 4;         // B: 16x128, 2 per thread
  const int br1  = (tid + 256) >> 5, bc1 = ((tid + 256) & 31) * 4;

  v8f acc00 = {}, acc01 = {}, acc10 = {}, acc11 = {};

  // Prologue: stage chunk 0 into buffer 0
  {
    async_b128(Wm + (size_t)(m0 + arow) * E_DIM + ac4, &At[0][arow][ac4]);
    async_b128(Xb + (size_t)br0 * W_DIM + n0 + bc0,    &Bt[0][br0][bc0]);
    async_b128(Xb + (size_t)br1 * W_DIM + n0 + bc1,    &Bt[0][br1][bc1]);
  }

  for (int c = 0; c < NCHUNK; ++c) {
    const int buf = c & 1;
    if (c + 1 < NCHUNK) {
      const int k1 = (c + 1) * 16;
      async_b128(Wm + (size_t)(m0 + arow) * E_DIM + k1 + ac4, &At[buf ^ 1][arow][ac4]);
      async_b128(Xb + (size_t)(k1 + br0) * W_DIM + n0 + bc0,  &Bt[buf ^ 1][br0][bc0]);
      async_b128(Xb + (size_t)(k1 + br1) * W_DIM + n0 + bc1,  &Bt[buf ^ 1][br1][bc1]);
      async_wait3();  // chunk c's 3 copies retired; prefetch may be in flight
    } else {
      async_wait0();
    }
    __syncthreads();

#pragma unroll
    for (int kk = 0; kk < 16; kk += 4) {
      const int ka = kk + hi * 2;  // half-wave k split per ISA f32 layouts
      v2f a0 = *(const v2f*)&At[buf][mw * 32 + hl][ka];
      v2f a1 = *(const v2f*)&At[buf][mw * 32 + 16 + hl][ka];
      v2f b0, b1;
      b0.x = Bt[buf][ka + 0][nw * 32 + hl];
      b0.y = Bt[buf][ka + 1][nw * 32 + hl];
      b1.x = Bt[buf][ka + 0][nw * 32 + 16 + hl];
      b1.y = Bt[buf][ka + 1][nw * 32 + 16 + hl];
      acc00 = wmma_f32(a0, b0, acc00);
      acc01 = wmma_f32(a0, b1, acc01);
      acc10 = wmma_f32(a1, b0, acc10);
      acc11 = wmma_f32(a1, b1, acc11);
    }
    __syncthreads();
  }

  // Epilogue: C-layout rows r and r+8 per half-wave, column = lane
  const int col0 = n0 + nw * 32 + hl;
  const int col1 = col0 + 16;
  for (int r = 0; r < 8; ++r) {
    int mr0 = m0 + mw * 32 + r + hi * 8;
    int mr1 = mr0 + 16;
    float ad0 = bias ? bias[mr0] : 0.0f;
    float ad1 = bias ? bias[mr1] : 0.0f;
    Cb[(size_t)mr0 * W_DIM + col0] = acc00[r] + ad0;
    Cb[(size_t)mr0 * W_DIM + col1] = acc01[r] + ad0;
    Cb[(size_t)mr1 * W_DIM + col0] = acc10[r] + ad1;
    Cb[(size_t)mr1 * W_DIM + col1] = acc11[r] + ad1;
  }
}

// ---------------------------------------------------------------------------
// Fused attention per (b, h, key-column block of 64).
// Score panel [qmax][64] lives entirely in LDS (<= 272KB of the 320KB WGP LDS).
//   Phase 1: S = Q^T K via WMMA (tiles -> LDS panel)
//   Phase 2: per-column masked softmax over q (q>k -> 0), * E^-0.5
//   Phase 3: O = V @ P via WMMA (B-fragments re-read from LDS panel)
// grid = (W/64, H, B), block = 256 threads
// ---------------------------------------------------------------------------
#define SP 68  // score panel pitch (floats)

__global__ __launch_bounds__(256)
void attn_kernel(const float* __restrict__ Qp, const float* __restrict__ Kp,
                 const float* __restrict__ Vp, float* __restrict__ Y) {
  extern __shared__ float smem[];
  float* Sp  = smem;                       // [1024][SP] score/prob panel
  float* Kt  = Sp + 1024 * SP;             // [64][SP]   staged K tile
  float* red = Kt + 64 * SP;               // [8][64]    softmax reductions

  const int tid  = threadIdx.x;
  const int lane = tid & 31;
  const int wave = tid >> 5;
  const int hl   = lane & 15;
  const int hi   = lane >> 4;

  const int k0 = blockIdx.x * 64;
  const int h  = blockIdx.y;
  const int b  = blockIdx.z;
  const size_t bo = (size_t)b * E_DIM * W_DIM + (size_t)h * DH_DIM * W_DIM;
  const float* Qb = Qp + bo;  // [64 d][W]
  const float* Kb = Kp + bo;
  const float* Vb = Vp + bo;
  float*       Yb = Y + bo;

  const int qmax = k0 + 64;  // rows (queries) needed for this key block

  // Stage K tile [64 d][64 cols] via async copies
  for (int j = 0; j < 4; ++j) {
    int v = tid + j * 256;          // float4 id 0..1023
    int d = v >> 4, cg = (v & 15) * 4;
    async_b128(Kb + (size_t)d * W_DIM + k0 + cg, Kt + d * SP + cg);
  }
  async_wait0();
  __syncthreads();

  // ---- Phase 1: scores -> Sp -------------------------------------------
  const int nt = wave & 3;  // key-column subtile
  for (int qt = (wave >> 2); qt < (qmax >> 4); qt += 2) {
    const int q0 = qt * 16;
    v8f acc = {};
#pragma unroll
    for (int d = 0; d < 64; d += 4) {
      const int da = d + hi * 2;
      v2f a;  // A[q][d] = Q[d][q] (transposed read, lane-coalesced over q)
      a.x = Qb[(size_t)da * W_DIM + q0 + hl];
      a.y = Qb[(size_t)(da + 1) * W_DIM + q0 + hl];
      v2f bb;
      bb.x = Kt[da * SP + nt * 16 + hl];
      bb.y = Kt[(da + 1) * SP + nt * 16 + hl];
      acc = wmma_f32(a, bb, acc);
    }
    for (int r = 0; r < 8; ++r)
      Sp[(q0 + r + hi * 8) * SP + nt * 16 + hl] = acc[r];
  }
  __syncthreads();

  // ---- Phase 2: masked softmax over q per column, * 1/32 ---------------
  {
    const float scale = 0.03125f;  // 1024^-0.5
    const int c     = tid & 63;    // column in block
    const int part  = tid >> 6;    // 4 threads per column
    const int kglob = k0 + c;
    const int chunk = qmax >> 2;
    const int rbeg = part * chunk, rend = rbeg + chunk;

    float m = -3.4e38f;
    for (int r = rbeg; r < rend; ++r)
      if (r <= kglob) m = fmaxf(m, Sp[r * SP + c]);
    red[part * 64 + c] = m;
    __syncthreads();
    m = fmaxf(fmaxf(red[0 * 64 + c], red[1 * 64 + c]),
              fmaxf(red[2 * 64 + c], red[3 * 64 + c]));

    float l = 0.0f;
    for (int r = rbeg; r < rend; ++r)
      if (r <= kglob) l += __expf(Sp[r * SP + c] - m);
    red[(4 + part) * 64 + c] = l;
    __syncthreads();
    l = red[4 * 64 + c] + red[5 * 64 + c] + red[6 * 64 + c] + red[7 * 64 + c];

    const float inv = scale / l;
    for (int r = rbeg; r < rend; ++r) {
      float p = (r <= kglob) ? __expf(Sp[r * SP + c] - m) * inv : 0.0f;
      Sp[r * SP + c] = p;
    }
  }
  __syncthreads();

  // ---- Phase 3: O = V @ P ----------------------------------------------
  const int mt0 = (wave >> 2) * 2;
  v8f o0 = {}, o1 = {};
  for (int q = 0; q < qmax; q += 4) {
    const int qa = q + hi * 2;
    v2f a0 = *(const v2f*)(Vb + (size_t)(mt0 * 16 + hl) * W_DIM + qa);
    v2f a1 = *(const v2f*)(Vb + (size_t)((mt0 + 1) * 16 + hl) * W_DIM + qa);
    v2f bb;
    bb.x = Sp[qa * SP + nt * 16 + hl];
    bb.y = Sp[(qa + 1) * SP + nt * 16 + hl];
    o0 = wmma_f32(a0, bb, o0);
    o1 = wmma_f32(a1, bb, o1);
  }
  const int ncol = k0 + nt * 16 + hl;
  for (int r = 0; r < 8; ++r) {
    Yb[(size_t)(mt0 * 16 + r + hi * 8) * W_DIM + ncol]       = o0[r];
    Yb[(size_t)((mt0 + 1) * 16 + r + hi * 8) * W_DIM + ncol] = o1[r];
  }
}

// ---------------------------------------------------------------------------
extern "C" void kernel_launch(void* const* d_in, const int* in_sizes, int n_in,
                              void* d_out, int out_size, void* d_ws, size_t ws_size,
                              hipStream_t stream) {
  (void)in_sizes; (void)n_in; (void)out_size; (void)ws_size;
  const float* x       = (const float*)d_in[0];  // [3, B, E, W]
  const float* L_Q     = (const float*)d_in[1];
  const float* L_K     = (const float*)d_in[2];
  const float* L_V     = (const float*)d_in[3];
  const float* dense_M = (const float*)d_in[4];
  const float* dense_b = (const float*)d_in[5];
  float* out = (float*)d_out;

  const size_t plane = (size_t)B_DIM * E_DIM * W_DIM;  // 8M floats
  float* Kp = (float*)d_ws;          // projected K
  float* Qp = Kp + plane;            // projected Q
  float* Vp = Qp + plane;            // projected V
  float* Yv = Vp + plane;            // attention output (pre-dense)

  dim3 blk(256);
  dim3 g1(W_DIM / 128, E_DIM / 64, B_DIM);
  // reference: K,Q,V = x[0],x[1],x[2]; K uses L_K, Q uses L_Q, V uses L_V
  gemm_f32_kernel<<<g1, blk, 0, stream>>>(L_K, x + 0 * plane, Kp, nullptr);
  gemm_f32_kernel<<<g1, blk, 0, stream>>>(L_Q, x + 1 * plane, Qp, nullptr);
  gemm_f32_kernel<<<g1, blk, 0, stream>>>(L_V, x + 2 * plane, Vp, nullptr);

  const size_t smem_bytes = (size_t)(1024 * SP + 64 * SP + 8 * 64) * sizeof(float);
  dim3 g2(W_DIM / 64, H_DIM, B_DIM);
  attn_kernel<<<g2, blk, smem_bytes, stream>>>(Qp, Kp, Vp, Yv);

  gemm_f32_kernel<<<g1, blk, 0, stream>>>(dense_M, Yv, out, dense_b);
}